// ECDALoss_33002528703277
// MI455X (gfx1250) — compile-verified
//
#include <hip/hip_runtime.h>
#include <math.h>

// ECDA loss for MI455X (gfx1250, wave32, WMMA).
// N=2048 points, D=256 dims, 8 classes, 5 Gaussian kernels.
// Two tiled WMMA-f32 passes over the three pairwise-distance GEMMs
// (cc, nn, cn), with per-class bucketed reductions via LDS+global atomics.
// Register-blocked 2x4: each wave computes a 32x64 region (8 accumulators)
// -> 0.75 global loads per v_wmma (f32 WMMA K=4 is operand-fetch bound).

#define NF 2048
#define DF 256
#define NC 8
#define KNUM 5
#define THRESH 0.5f
#define LAMBDA 1.0f
#define GAMMA  0.1f
#define DELTA  0.01f

// workspace layout (float indices)
#define OFF_NORMC   0        // [2048] clean row sq-norms
#define OFF_NORMN   2048     // [2048] noisy row sq-norms
#define OFF_CEN     4096     // [8*256] class centroids
#define OFF_NS      6144     // [8] clean counts
#define OFF_NT      6152     // [8] masked noisy counts
#define OFF_SW      6160     // [8] sum of masked noisy scores
#define OFF_SUML2   6168     // [8] per-class sum of d^2 (atomic)
#define OFF_BW0     6176     // [8] base bandwidth
#define OFF_SS      6184     // [8] kernel sums clean-clean (atomic)
#define OFF_TT      6192     // [8] kernel sums noisy-noisy (atomic)
#define OFF_ST      6200     // [8] kernel sums clean-noisy (atomic)
#define OFF_COMPACT 6208     // [8] raw compactness sums
#define OFF_CLSN    6224     // int[2048]: noisy class or -1 if mask<=thresh
#define OFF_WN      8272     // [2048] masked noisy score weights

typedef __attribute__((ext_vector_type(2))) float v2f;
typedef __attribute__((ext_vector_type(8))) float v8f;

// ---------------------------------------------------------------- zero stats
__global__ void k_zero(float* __restrict__ W) {
    for (int i = OFF_CEN + threadIdx.x; i < OFF_CLSN; i += 256) W[i] = 0.f;
}

// ------------------------------------------- row sq-norms + derived noisy arrays
__global__ void k_rows(const float* __restrict__ cF, const float* __restrict__ nF,
                       const int* __restrict__ nL, const float* __restrict__ mask,
                       const float* __restrict__ sc, float* __restrict__ W) {
    int wave = threadIdx.x >> 5, lane = threadIdx.x & 31;
    int r = blockIdx.x * 8 + wave;               // 0..4095: clean rows then noisy rows
    const float* src = (r < NF) ? (cF + (size_t)r * DF) : (nF + (size_t)(r - NF) * DF);
    float s = 0.f;
    #pragma unroll
    for (int d = lane; d < DF; d += 32) { float x = src[d]; s += x * x; }
    #pragma unroll
    for (int o = 16; o; o >>= 1) s += __shfl_xor(s, o, 32);
    if (lane == 0) {
        if (r < NF) {
            W[OFF_NORMC + r] = s;
        } else {
            int j = r - NF;
            W[OFF_NORMN + j] = s;
            bool ok = mask[j] > THRESH;
            ((int*)W)[OFF_CLSN + j] = ok ? nL[j] : -1;
            W[OFF_WN + j] = ok ? sc[j] : 0.f;
        }
    }
}

// -------------------------------- per-class counts, score sums, centroids
__global__ void k_classstats(const float* __restrict__ nF, const int* __restrict__ cL,
                             float* __restrict__ W) {
    int c = blockIdx.x, tid = threadIdx.x;       // tid = feature dim
    const int* clsN = ((const int*)W) + OFF_CLSN;
    __shared__ float s_cnt;
    float acc = 0.f;
    for (int j = 0; j < NF; ++j)
        if (clsN[j] == c) acc += nF[(size_t)j * DF + tid];
    if (tid == 0) {
        float cnt = 0.f, swv = 0.f;
        for (int j = 0; j < NF; ++j)
            if (clsN[j] == c) { cnt += 1.f; swv += W[OFF_WN + j]; }
        s_cnt = cnt; W[OFF_NT + c] = cnt; W[OFF_SW + c] = swv;
    }
    if (tid == 1) {
        float ns = 0.f;
        for (int i = 0; i < NF; ++i) ns += (cL[i] == c) ? 1.f : 0.f;
        W[OFF_NS + c] = ns;
    }
    __syncthreads();
    W[OFF_CEN + c * DF + tid] = acc / fmaxf(s_cnt, 1.f);
}

// -------------------------------------- WMMA 32x64 register-blocked tile
// One wave computes a 32x64 f32 region of A[rowA..+31] dot B[rowB..+63] as
// 2x4 tiles of V_WMMA_F32_16X16X4_F32 over K=256 in steps of 4.
// For A.B^T with both operands row-major, A and B fragments have identical
// per-lane loads: lane L<16 -> row L cols {k,k+1}; L>=16 -> row L-16 cols
// {k+2,k+3}. 6 loads feed 8 WMMAs per step (0.75 load/wmma), and the 8
// independent accumulator chains hide the WMMA->WMMA RAW latency.
__device__ __forceinline__ void wmma_dot_tile(const float* __restrict__ A,
                                              const float* __restrict__ B,
                                              int rowA, int rowB, int lane,
                                              v8f acc[2][4]) {
    int l16 = lane & 15, hi = lane >> 4;
    const float* ap0 = A + (size_t)(rowA + l16) * DF + 2 * hi;
    const float* ap1 = ap0 + 16 * DF;
    const float* bp0 = B + (size_t)(rowB + l16) * DF + 2 * hi;
    const float* bp1 = bp0 + 16 * DF;
    const float* bp2 = bp0 + 32 * DF;
    const float* bp3 = bp0 + 48 * DF;
    #pragma unroll
    for (int ti = 0; ti < 2; ++ti)
        #pragma unroll
        for (int tj = 0; tj < 4; ++tj)
            acc[ti][tj] = v8f{};
    #pragma unroll 2
    for (int k = 0; k < DF; k += 4) {
        v2f a0 = *(const v2f*)(ap0 + k);
        v2f a1 = *(const v2f*)(ap1 + k);
        v2f b0 = *(const v2f*)(bp0 + k);
        v2f b1 = *(const v2f*)(bp1 + k);
        v2f b2 = *(const v2f*)(bp2 + k);
        v2f b3 = *(const v2f*)(bp3 + k);
        acc[0][0] = __builtin_amdgcn_wmma_f32_16x16x4_f32(false, a0, false, b0,
                                                          (short)0, acc[0][0], false, false);
        acc[0][1] = __builtin_amdgcn_wmma_f32_16x16x4_f32(false, a0, false, b1,
                                                          (short)0, acc[0][1], false, false);
        acc[0][2] = __builtin_amdgcn_wmma_f32_16x16x4_f32(false, a0, false, b2,
                                                          (short)0, acc[0][2], false, false);
        acc[0][3] = __builtin_amdgcn_wmma_f32_16x16x4_f32(false, a0, false, b3,
                                                          (short)0, acc[0][3], false, false);
        acc[1][0] = __builtin_amdgcn_wmma_f32_16x16x4_f32(false, a1, false, b0,
                                                          (short)0, acc[1][0], false, false);
        acc[1][1] = __builtin_amdgcn_wmma_f32_16x16x4_f32(false, a1, false, b1,
                                                          (short)0, acc[1][1], false, false);
        acc[1][2] = __builtin_amdgcn_wmma_f32_16x16x4_f32(false, a1, false, b2,
                                                          (short)0, acc[1][2], false, false);
        acc[1][3] = __builtin_amdgcn_wmma_f32_16x16x4_f32(false, a1, false, b3,
                                                          (short)0, acc[1][3], false, false);
    }
}

// ---------------------------------------- pass 1: per-class sum of d^2 (bandwidth)
// grid: (16, 16, 3) x 256 threads. z: 0=cc, 1=nn, 2=cn.
// 8 waves (4x2) of 32x64 regions -> 128x128 block region.
__global__ void k_pass1(const float* __restrict__ cF, const float* __restrict__ nF,
                        const int* __restrict__ cL, float* __restrict__ W) {
    int z = blockIdx.z;
    const float* A    = (z == 1) ? nF : cF;
    const float* B    = (z == 0) ? cF : nF;
    const float* nrmA = W + ((z == 1) ? OFF_NORMN : OFF_NORMC);
    const float* nrmB = W + ((z == 0) ? OFF_NORMC : OFF_NORMN);
    const int*   clsA = (z == 1) ? (((const int*)W) + OFF_CLSN) : cL;
    const int*   clsB = (z == 0) ? cL : (((const int*)W) + OFF_CLSN);
    float wz = (z == 2) ? 2.f : 1.f;             // cn term counts twice in sumL2

    __shared__ float s_sum[NC];
    if (threadIdx.x < NC) s_sum[threadIdx.x] = 0.f;
    __syncthreads();

    int wave = threadIdx.x >> 5, lane = threadIdx.x & 31;
    int rowA = blockIdx.y * 128 + (wave >> 1) * 32;
    int rowB = blockIdx.x * 128 + (wave & 1) * 64;

    v8f acc[2][4];
    wmma_dot_tile(A, B, rowA, rowB, lane, acc);

    int l16 = lane & 15, hi = lane >> 4;
    #pragma unroll
    for (int tj = 0; tj < 4; ++tj) {
        int j = rowB + tj * 16 + l16;
        int cj = clsB[j];
        if (cj < 0) continue;
        float nb = nrmB[j];
        #pragma unroll
        for (int ti = 0; ti < 2; ++ti) {
            int ibase = rowA + ti * 16 + 8 * hi;  // C/D layout: M = v + 8*(lane>=16)
            #pragma unroll
            for (int v = 0; v < 8; ++v) {
                int i = ibase + v;
                if (clsA[i] == cj) {
                    float d2 = fmaxf(nrmA[i] + nb - 2.f * acc[ti][tj][v], 0.f);
                    atomicAdd(&s_sum[cj], wz * d2);
                }
            }
        }
    }
    __syncthreads();
    if (threadIdx.x < NC) atomicAdd(&W[OFF_SUML2 + threadIdx.x], s_sum[threadIdx.x]);
}

// --------------------------------------------------------------- bandwidths
__global__ void k_bw0(float* __restrict__ W) {
    int c = threadIdx.x;
    if (c < NC) {
        float n = W[OFF_NS + c] + W[OFF_NT + c];
        W[OFF_BW0 + c] = W[OFF_SUML2 + c] / fmaxf(n * n - n, 1.f) / 4.0f; // half=2^(5/2)=4
    }
}

// -------------------------------------------------------- compactness sums
__global__ void k_compact(const float* __restrict__ nF, float* __restrict__ W) {
    int c = blockIdx.x, tid = threadIdx.x;
    const int* clsN = ((const int*)W) + OFF_CLSN;
    float cen = W[OFF_CEN + c * DF + tid];
    float acc = 0.f;
    for (int j = 0; j < NF; ++j)
        if (clsN[j] == c) { float d = nF[(size_t)j * DF + tid] - cen; acc += d * d; }
    __shared__ float s_red[256];
    s_red[tid] = acc; __syncthreads();
    #pragma unroll
    for (int s = 128; s; s >>= 1) { if (tid < s) s_red[tid] += s_red[tid + s]; __syncthreads(); }
    if (tid == 0) W[OFF_COMPACT + c] = s_red[0];
}

// ------------------------------- pass 2: multi-bandwidth Gaussian kernel sums
__global__ void k_pass2(const float* __restrict__ cF, const float* __restrict__ nF,
                        const int* __restrict__ cL, float* __restrict__ W) {
    int z = blockIdx.z;
    const float* A    = (z == 1) ? nF : cF;
    const float* B    = (z == 0) ? cF : nF;
    const float* nrmA = W + ((z == 1) ? OFF_NORMN : OFF_NORMC);
    const float* nrmB = W + ((z == 0) ? OFF_NORMC : OFF_NORMN);
    const int*   clsA = (z == 1) ? (((const int*)W) + OFF_CLSN) : cL;
    const int*   clsB = (z == 0) ? cL : (((const int*)W) + OFF_CLSN);
    const float* wn   = W + OFF_WN;

    __shared__ float s_sum[NC];
    __shared__ float s_ninv[NC * KNUM];          // -1/(bw0*2^k + 1e-8)
    if (threadIdx.x < NC) s_sum[threadIdx.x] = 0.f;
    if (threadIdx.x < NC * KNUM) {
        int c = threadIdx.x / KNUM, k = threadIdx.x % KNUM;
        s_ninv[threadIdx.x] = -1.f / (W[OFF_BW0 + c] * (float)(1 << k) + 1e-8f);
    }
    __syncthreads();

    int wave = threadIdx.x >> 5, lane = threadIdx.x & 31;
    int rowA = blockIdx.y * 128 + (wave >> 1) * 32;
    int rowB = blockIdx.x * 128 + (wave & 1) * 64;

    v8f acc[2][4];
    wmma_dot_tile(A, B, rowA, rowB, lane, acc);

    int l16 = lane & 15, hi = lane >> 4;
    #pragma unroll
    for (int tj = 0; tj < 4; ++tj) {
        int j = rowB + tj * 16 + l16;
        int cj = clsB[j];
        if (cj < 0) continue;
        float nb = nrmB[j];
        float wj = (z == 0) ? 1.f : wn[j];
        const float* pk = &s_ninv[cj * KNUM];
        #pragma unroll
        for (int ti = 0; ti < 2; ++ti) {
            int ibase = rowA + ti * 16 + 8 * hi;
            #pragma unroll
            for (int v = 0; v < 8; ++v) {
                int i = ibase + v;
                if (clsA[i] == cj) {
                    float d2 = fmaxf(nrmA[i] + nb - 2.f * acc[ti][tj][v], 0.f);
                    float s = __expf(d2 * pk[0]) + __expf(d2 * pk[1]) + __expf(d2 * pk[2])
                            + __expf(d2 * pk[3]) + __expf(d2 * pk[4]);
                    float wi = (z == 1) ? wn[i] : 1.f;
                    atomicAdd(&s_sum[cj], wi * wj * s);
                }
            }
        }
    }
    __syncthreads();
    if (threadIdx.x < NC) {
        float* dst = W + ((z == 0) ? OFF_SS : (z == 1) ? OFF_TT : OFF_ST);
        atomicAdd(&dst[threadIdx.x], s_sum[threadIdx.x]);
    }
}

// ------------------------------------------------ repulsion + MMD + total
__global__ void k_final(const float* __restrict__ wce, float* __restrict__ W,
                        float* __restrict__ out) {
    __shared__ float s_sumD, s_npair, s_nvalid;
    int t = threadIdx.x;
    if (t == 0) { s_sumD = 0.f; s_npair = 0.f; s_nvalid = 0.f; }
    __syncthreads();
    if (t < 64) {
        int a = t >> 3, b = t & 7;
        if (a < b && W[OFF_NT + a] > 0.f && W[OFF_NT + b] > 0.f) {
            float d2 = 0.f;
            for (int d = 0; d < DF; ++d) {
                float df = W[OFF_CEN + a * DF + d] - W[OFF_CEN + b * DF + d];
                d2 += df * df;
            }
            atomicAdd(&s_sumD, sqrtf(fmaxf(d2, 1e-12f)));
            atomicAdd(&s_npair, 1.f);
        }
        if (b == 0 && W[OFF_NT + a] > 0.f) atomicAdd(&s_nvalid, 1.f);
    }
    __syncthreads();
    if (t == 0) {
        float repulsion = (s_nvalid > 1.f) ? (-s_sumD / fmaxf(s_npair, 1.f)) : 0.f;
        float mean = 0.f;
        for (int c = 0; c < NC; ++c) mean += wce[c];
        mean *= (1.f / NC);
        float total = 0.f;
        for (int c = 0; c < NC; ++c) {
            float ns = W[OFF_NS + c], nt = W[OFF_NT + c], sw = W[OFF_SW + c];
            float mmd = W[OFF_SS + c] / (ns * ns + 1e-8f)
                      + W[OFF_TT + c] / (sw * sw + 1e-8f)
                      - 2.f * W[OFF_ST + c] / (ns * sw + 1e-8f);
            float compact = W[OFF_COMPACT + c] / fmaxf(nt, 1.f);
            float attn = __expf(LAMBDA * (mean - wce[c]));
            if (ns >= 2.f && nt >= 2.f)
                total += attn * (mmd + GAMMA * compact + DELTA * repulsion);
        }
        out[0] = total;
    }
}

extern "C" void kernel_launch(void* const* d_in, const int* in_sizes, int n_in,
                              void* d_out, int out_size, void* d_ws, size_t ws_size,
                              hipStream_t stream) {
    const float* cF   = (const float*)d_in[0];
    const float* nF   = (const float*)d_in[1];
    const int*   cL   = (const int*)d_in[2];
    const int*   nL   = (const int*)d_in[3];
    const float* mask = (const float*)d_in[4];
    const float* sc   = (const float*)d_in[5];
    const float* wce  = (const float*)d_in[6];
    float* W   = (float*)d_ws;
    float* out = (float*)d_out;

    k_zero<<<1, 256, 0, stream>>>(W);
    k_rows<<<512, 256, 0, stream>>>(cF, nF, nL, mask, sc, W);
    k_classstats<<<NC, 256, 0, stream>>>(nF, cL, W);
    dim3 g(NF / 128, NF / 128, 3);
    k_pass1<<<g, 256, 0, stream>>>(cF, nF, cL, W);
    k_bw0<<<1, 64, 0, stream>>>(W);
    k_compact<<<NC, 256, 0, stream>>>(nF, W);
    k_pass2<<<g, 256, 0, stream>>>(cF, nF, cL, W);
    k_final<<<1, 64, 0, stream>>>(wce, W, out);
}